// RotationAnisotropyLoss_35338990911618
// MI455X (gfx1250) — compile-verified
//
#include <hip/hip_runtime.h>
#include <math.h>

typedef __attribute__((ext_vector_type(2))) float v2f;
typedef __attribute__((ext_vector_type(8))) float v8f;

#define NB 8
#define NG 262144
#define GSTRIDE 14
#define WPB 8              // waves per block (256 threads)
#define GRIDX 128          // blocks per batch -> 1024 waves/batch, 8 iters/wave

__device__ __forceinline__ float sanitize_f(float v, float clampAbs) {
    v = (v != v) ? 0.0f : v;                     // NaN -> 0
    return fminf(fmaxf(v, -clampAbs), clampAbs); // (+-inf clipped) -> v_med3
}

__global__ void init_ws_kernel(float* __restrict__ ws) {
    int i = threadIdx.x;
    if (i < NB * 9) ws[i] = 0.0f;
}

// T[b] = sum_n r_n r_n^T via V_WMMA_F32_16X16X4_F32 rank-4 updates.
// Per wave-iteration: 32 lanes each compute r for one gaussian, stage r into a
// per-wave LDS strip, then 8 WMMAs (K=4 gaussians each) consume the strip.
// A(16x4) and B(4x16) register images are identical here (lane%16 = component
// row/col index, VGPR slot + lane-half = K), so one v2f feeds both operands.
__global__ void __launch_bounds__(256)
scatter_wmma_kernel(const float* __restrict__ g, float* __restrict__ ws) {
    __shared__ float lds_r[WPB * 96];  // 32 gaussians * 3 comps per wave
    __shared__ float lds_zero[96];     // zero pad for component lanes c >= 3
    __shared__ float ldsT[9];

    const int tid  = threadIdx.x;
    if (tid < 96) lds_zero[tid] = 0.0f;
    if (tid < 9)  ldsT[tid] = 0.0f;
    __syncthreads();

    const int batch = blockIdx.y;
    const int lane  = tid & 31;
    const int wave  = tid >> 5;
    const int c     = lane & 15;   // operand component index (valid 0..2)
    const int hs    = lane >> 4;   // lane half: K offset 0 / 2

    const float* gb = g + (size_t)batch * NG * GSTRIDE;

    // wave-uniform scalar loop bounds (keeps EXEC untouched around WMMA)
    const int w        = __builtin_amdgcn_readfirstlane(blockIdx.x * WPB + wave);
    const int stride32 = GRIDX * WPB * 32;

    float* wp = &lds_r[wave * 96 + lane * 3];
    const float* rp = (c < 3) ? &lds_r[wave * 96 + 2 * hs * 3 + c] : &lds_zero[0];

    v8f acc = {};
    for (int base = w * 32; base < NG; base += stride32) {
        // ---- compute r = R(q) @ e_z for this lane's gaussian ----
        const float* q = gb + (size_t)(base + lane) * GSTRIDE + 7;
        float qw = sanitize_f(q[0], 1000.0f);
        float qx = sanitize_f(q[1], 1000.0f);
        float qy = sanitize_f(q[2], 1000.0f);
        float qz = sanitize_f(q[3], 1000.0f);
        // 1/max(|q|,1e-8)^2 == rcp(max(|q|^2,1e-16)); hardware v_rcp_f32
        float n2  = qw*qw + qx*qx + qy*qy + qz*qz;
        float s   = 2.0f * __builtin_amdgcn_rcpf(fmaxf(n2, 1e-16f));
        float r0  = s * (qx*qz + qw*qy);
        float r1  = s * (qy*qz - qw*qx);
        float r2  = 1.0f - s * (qx*qx + qy*qy);

        // ---- stage into per-wave LDS strip ----
        wp[0] = r0; wp[1] = r1; wp[2] = r2;
        asm volatile("s_wait_dscnt 0" ::: "memory");  // in-order DS pipe + explicit wait

        // ---- 8 rank-4 WMMA updates consume the 32 staged gaussians ----
        #pragma unroll
        for (int t = 0; t < 8; ++t) {
            v2f a;
            a[0] = rp[t * 12 + 0];  // K = 0 (lanes 0-15) / K = 2 (lanes 16-31)
            a[1] = rp[t * 12 + 3];  // K = 1 / K = 3
            acc = __builtin_amdgcn_wmma_f32_16x16x4_f32(
                false, a, false, a, (short)0, acc, false, false);
        }
        asm volatile("" ::: "memory");  // keep next iter's stores below these loads
    }

    // C/D layout: lanes 0-15 hold D[M=v][N=lane] in VGPR v; only M,N in 0..2 nonzero.
    if (lane < 3) {
        atomicAdd(&ldsT[0 * 3 + lane], acc[0]);
        atomicAdd(&ldsT[1 * 3 + lane], acc[1]);
        atomicAdd(&ldsT[2 * 3 + lane], acc[2]);
    }
    __syncthreads();
    if (tid < 9) atomicAdd(&ws[batch * 9 + tid], ldsT[tid]);
}

// Closed-form symmetric 3x3 eigensolve + loss (8 batches, 1 thread, fp64)
__global__ void finalize_kernel(const float* __restrict__ ws, float* __restrict__ out) {
    if (threadIdx.x != 0 || blockIdx.x != 0) return;
    double loss = 0.0;
    for (int b = 0; b < NB; ++b) {
        double T[9];
        for (int i = 0; i < 9; ++i) {
            float v = sanitize_f(ws[b * 9 + i] / (float)NG, 1e6f);
            T[i] = (double)v;
        }
        double a00 = T[0], a11 = T[4], a22 = T[8];
        double a01 = 0.5 * (T[1] + T[3]);
        double a02 = 0.5 * (T[2] + T[6]);
        double a12 = 0.5 * (T[5] + T[7]);

        double p1 = a01 * a01 + a02 * a02 + a12 * a12;
        double qm = (a00 + a11 + a22) / 3.0;
        double emin, emax;
        if (p1 < 1e-30) {
            emin = fmin(a00, fmin(a11, a22));
            emax = fmax(a00, fmax(a11, a22));
        } else {
            double b00 = a00 - qm, b11 = a11 - qm, b22 = a22 - qm;
            double p2 = b00 * b00 + b11 * b11 + b22 * b22 + 2.0 * p1;
            double p  = sqrt(p2 / 6.0);
            double ip = 1.0 / p;
            double c00 = b00 * ip, c11 = b11 * ip, c22 = b22 * ip;
            double c01 = a01 * ip, c02 = a02 * ip, c12 = a12 * ip;
            double detB = c00 * (c11 * c22 - c12 * c12)
                        - c01 * (c01 * c22 - c12 * c02)
                        + c02 * (c01 * c12 - c11 * c02);
            double r = fmin(fmax(0.5 * detB, -1.0), 1.0);
            double phi = acos(r) / 3.0;
            emax = qm + 2.0 * p * cos(phi);
            emin = qm + 2.0 * p * cos(phi + 2.0943951023931953); // +2*pi/3
        }
        double maxs = fmax(fmin(fmax(emax, -1e6), 1e6), 1e-6);
        double mins = fmax(fmin(fmax(emin, -1e6), 1e6), 1e-6);
        double ratio = fmin(fmax(maxs / mins, 1.0), 1e6);
        loss += -log(ratio);
    }
    out[0] = (float)(loss / (double)NB);
}

extern "C" void kernel_launch(void* const* d_in, const int* in_sizes, int n_in,
                              void* d_out, int out_size, void* d_ws, size_t ws_size,
                              hipStream_t stream) {
    const float* gaussians = (const float*)d_in[0];
    float* out = (float*)d_out;
    float* ws  = (float*)d_ws;  // 72 floats: 8 batches x 3x3 accumulators

    init_ws_kernel<<<1, 128, 0, stream>>>(ws);

    dim3 grid(GRIDX, NB, 1);   // 1024 blocks x 8 waves; 8 iterations/wave
    dim3 block(256, 1, 1);
    scatter_wmma_kernel<<<grid, block, 0, stream>>>(gaussians, ws);

    finalize_kernel<<<1, 32, 0, stream>>>(ws, out);
}